// CMCMem_90632399880357
// MI455X (gfx1250) — compile-verified
//
#include <hip/hip_runtime.h>
#include <math.h>

typedef float v2f __attribute__((ext_vector_type(2)));
typedef float v4f __attribute__((ext_vector_type(4)));
typedef float v8f __attribute__((ext_vector_type(8)));

// Problem constants (from the reference)
constexpr int B   = 128;
constexpr int D   = 128;
constexpr int N   = 600000;
constexpr int K   = 4096;
constexpr int KP1 = K + 1;                  // 4097
constexpr float INV_T = 1.0f / 0.07f;

// d_out layout (floats): out_l | out_ab | new_memory_l | new_memory_ab
constexpr long long OFF_OUT_L  = 0;
constexpr long long OFF_OUT_AB = (long long)B * KP1;             // 524,416
constexpr long long OFF_MEM_L  = 2LL * B * KP1;                  // 1,048,832
constexpr long long OFF_MEM_AB = OFF_MEM_L + (long long)N * D;   // 77,848,832

constexpr int WAVES_PER_BLOCK = 4;
constexpr int TILES_PER_B     = (KP1 + 15) / 16;                               // 257
constexpr int TILE_GROUPS     = (TILES_PER_B + WAVES_PER_BLOCK - 1) / WAVES_PER_BLOCK; // 65
constexpr int LDS_STRIDE      = 132;  // 128 + 4 pad: WMMA A-frag b64 reads hit all 64 banks once

// ---------------------------------------------------------------------------
// ws[0] = sum(exp) for out_ab, ws[1] = sum(exp) for out_l
__global__ void cmc_zero(float* ws) {
    if (threadIdx.x < 2) ws[threadIdx.x] = 0.0f;
}

// ---------------------------------------------------------------------------
// Streaming copy of both memory banks into d_out (non-temporal stores: the
// destinations are never re-read; sources stay cacheable for the gather pass).
__global__ void cmc_copy(const v4f* __restrict__ sl, const v4f* __restrict__ sab,
                         v4f* __restrict__ dl, v4f* __restrict__ dab, int n4) {
    int i = blockIdx.x * blockDim.x + threadIdx.x;
    if (i < n4) {
        __builtin_nontemporal_store(sl[i],  &dl[i]);
        __builtin_nontemporal_store(sab[i], &dab[i]);
    }
}

// ---------------------------------------------------------------------------
// Gather + dot + exp via V_WMMA_F32_16X16X4_F32 (exact f32 math).
// One wave per 16-k tile; B-matrix columns are the broadcast query chunk, so
// every column of D equals the 16 logits for this tile.
__global__ __launch_bounds__(WAVES_PER_BLOCK * 32) void cmc_logits(
    const float* __restrict__ l, const float* __restrict__ ab,
    const int* __restrict__ idx,
    const float* __restrict__ mem_l, const float* __restrict__ mem_ab,
    float* __restrict__ out, float* __restrict__ ws)
{
    __shared__ float tile[WAVES_PER_BLOCK][16 * LDS_STRIDE];
    __shared__ float qvec[WAVES_PER_BLOCK][D];

    const int w    = threadIdx.x >> 5;
    const int lane = threadIdx.x & 31;
    const int b    = blockIdx.y;
    const int bank = blockIdx.z;   // 0: memory_l · ab -> out_ab ; 1: memory_ab · l -> out_l

    const float* memv = bank ? mem_ab : mem_l;
    const float* qv   = bank ? l : ab;
    float* outp = out + (bank ? OFF_OUT_L : OFF_OUT_AB) + (long long)b * KP1;

    // Stage the 128-float query vector (wave-private copy: no cross-wave sync).
    {
        float4 v = ((const float4*)(qv + b * D))[lane];
        *(float4*)&qvec[w][lane * 4] = v;
    }

    const int k0 = (blockIdx.x * WAVES_PER_BLOCK + w) * 16;
    if (k0 >= KP1) return;   // wave-uniform exit (EXEC stays all-1s for WMMA)

    // Stage 16 gathered rows: one row per iteration = 32 lanes x float4 = 512B coalesced.
    const int* idxb = idx + b * KP1;
    for (int r = 0; r < 16; ++r) {
        int k  = k0 + r;
        int kk = (k < KP1) ? k : k0;            // clamp tail to a valid row
        int row = idxb[kk];
        float4 v = ((const float4*)(memv + (size_t)row * D))[lane];
        *(float4*)&tile[w][r * LDS_STRIDE + lane * 4] = v;
    }

    // WMMA accumulation over 32 chunks of K=4.
    // A frag: lanes 0-15 -> row=lane cols {c,c+1}; lanes 16-31 -> row=lane-16 cols {c+2,c+3}.
    // B frag (broadcast columns): same K mapping applied to the query vector.
    const int off = (lane < 16) ? 0 : 2;
    const int rb  = (lane < 16) ? lane : (lane - 16);
    const float* ta = &tile[w][rb * LDS_STRIDE + off];
    const float* tb = &qvec[w][off];

    v8f acc = {0.f, 0.f, 0.f, 0.f, 0.f, 0.f, 0.f, 0.f};
#pragma unroll
    for (int c = 0; c < D; c += 4) {
        v2f afrag = *(const v2f*)(ta + c);
        v2f bfrag = *(const v2f*)(tb + c);
        acc = __builtin_amdgcn_wmma_f32_16x16x4_f32(false, afrag, false, bfrag,
                                                    (short)0, acc, false, false);
    }

    // D layout: VGPR r = row r (lanes 0-15) / row r+8 (lanes 16-31); all columns equal.
    if (lane == 0 || lane == 16) {
        int base = k0 + ((lane == 16) ? 8 : 0);
        float s = 0.0f;
#pragma unroll
        for (int r = 0; r < 8; ++r) {
            int k = base + r;
            if (k < KP1) {
                float e = __expf(acc[r] * INV_T);
                outp[k] = e;
                s += e;
            }
        }
        atomicAdd(&ws[bank], s);
    }
}

// ---------------------------------------------------------------------------
// out /= Z, Z = sum/(B*(K+1)) * N
__global__ void cmc_scale(float* __restrict__ out, const float* __restrict__ ws) {
    long long i = (long long)blockIdx.x * blockDim.x + threadIdx.x;
    const long long half = (long long)B * KP1;
    if (i >= 2 * half) return;
    float sum = ws[(i < half) ? 1 : 0];          // out_l first -> ws[1]; out_ab -> ws[0]
    float scale = (float)((double)half / ((double)N * (double)sum));
    out[i] *= scale;
}

// ---------------------------------------------------------------------------
// Momentum update + L2 renorm of the 128 touched rows (runs after cmc_copy).
__global__ __launch_bounds__(32) void cmc_update(
    const float* __restrict__ l, const float* __restrict__ ab,
    const int* __restrict__ y,
    const float* __restrict__ mem_l, const float* __restrict__ mem_ab,
    float* __restrict__ out)
{
    const int b = blockIdx.x, bank = blockIdx.y, lane = threadIdx.x;
    const float* memv = bank ? mem_ab : mem_l;
    const float* vv   = bank ? ab : l;           // memory_l pairs with l, memory_ab with ab
    float* dst = out + (bank ? OFF_MEM_AB : OFF_MEM_L);

    const int row = y[b];
    float4 m4 = ((const float4*)(memv + (size_t)row * D))[lane];
    float4 v4 = ((const float4*)(vv + b * D))[lane];
    float4 p;
    p.x = m4.x * 0.5f + v4.x * 0.5f;
    p.y = m4.y * 0.5f + v4.y * 0.5f;
    p.z = m4.z * 0.5f + v4.z * 0.5f;
    p.w = m4.w * 0.5f + v4.w * 0.5f;

    float ss = p.x * p.x + p.y * p.y + p.z * p.z + p.w * p.w;
#pragma unroll
    for (int o = 16; o > 0; o >>= 1) ss += __shfl_xor(ss, o, 32);
    float rn = 1.0f / sqrtf(ss);
    p.x *= rn; p.y *= rn; p.z *= rn; p.w *= rn;
    ((float4*)(dst + (size_t)row * D))[lane] = p;
}

// ---------------------------------------------------------------------------
extern "C" void kernel_launch(void* const* d_in, const int* in_sizes, int n_in,
                              void* d_out, int out_size, void* d_ws, size_t ws_size,
                              hipStream_t stream)
{
    const float* l      = (const float*)d_in[0];
    const float* ab     = (const float*)d_in[1];
    const int*   y      = (const int*)d_in[2];
    const int*   idx    = (const int*)d_in[3];
    const float* mem_l  = (const float*)d_in[4];
    const float* mem_ab = (const float*)d_in[5];
    float* out = (float*)d_out;
    float* ws  = (float*)d_ws;

    cmc_zero<<<1, 32, 0, stream>>>(ws);

    const int n4 = N * D / 4;                       // 19,200,000 float4 per bank
    cmc_copy<<<(n4 + 255) / 256, 256, 0, stream>>>(
        (const v4f*)mem_l, (const v4f*)mem_ab,
        (v4f*)(out + OFF_MEM_L), (v4f*)(out + OFF_MEM_AB), n4);

    dim3 lg(TILE_GROUPS, B, 2);
    cmc_logits<<<lg, WAVES_PER_BLOCK * 32, 0, stream>>>(l, ab, idx, mem_l, mem_ab, out, ws);

    const long long tot = 2LL * B * KP1;
    cmc_scale<<<(int)((tot + 255) / 256), 256, 0, stream>>>(out, ws);

    dim3 ug(B, 2);
    cmc_update<<<ug, 32, 0, stream>>>(l, ab, y, mem_l, mem_ab, out);
}